// GAT_70050916598070
// MI455X (gfx1250) — compile-verified
//
#include <hip/hip_runtime.h>
#include <cstdint>
#include <cstddef>

// ---------------- GAT, 2 layers, gfx1250 (MI455X) ----------------
// Roofline: ~0.5 GFLOP total; cost is 6 edge passes (gather + L2 atomics).
// Layer1 (in=1, H=8, C=16) algebraically collapsed:
//   h1[n,h,c] = x[n]*W1[h,c]  =>  logits are x[n]*s1[h], x[n]*d1[h];
//   aggregation: out1[n,h,c] = W1[h,c]*t1[n,h], t1[n,h] = sum_e alpha*x[src]
//   -> 8 atomics/edge instead of 128; z1 never materialized.
// Layer2 (in=128, H=1, C=1) is a scalar GAT over h2[n] = elu(z1[n,:]) @ W2,
// computed with V_WMMA_F32_16X16X4_F32 (exact f32) fused with ELU, operands
// staged through LDS.
// Self-loops are a +N thread tail on each edge pass. edge_index (int64) is
// converted once to int32 in workspace to halve per-pass streaming traffic.

#define NHEAD 8
#define NEG_SLOPE 0.2f
#define EPS_DEN 1e-16f

typedef __attribute__((ext_vector_type(2))) float v2f;
typedef __attribute__((ext_vector_type(8))) float v8f;

__device__ __forceinline__ float lrelu(float v) { return v > 0.f ? v : NEG_SLOPE * v; }
__device__ __forceinline__ float eluf(float v)  { return v > 0.f ? v : expm1f(v); }

// order-preserving float<->uint map so atomicMax(u32) == float max
__device__ __forceinline__ unsigned ordf(float f) {
  unsigned b = __float_as_uint(f);
  return (b & 0x80000000u) ? ~b : (b | 0x80000000u);
}
__device__ __forceinline__ float unordf(unsigned u) {
  unsigned b = (u & 0x80000000u) ? (u ^ 0x80000000u) : ~u;
  return __uint_as_float(b);
}

// ---------------- utility kernels ----------------

__global__ void zero_k(unsigned* __restrict__ p, size_t n) {
  size_t t = (size_t)blockIdx.x * blockDim.x + threadIdx.x;
  if (t < n) p[t] = 0u;
}

__global__ void cvt_k(const long long* __restrict__ ei, int* __restrict__ e32, size_t n) {
  size_t t = (size_t)blockIdx.x * blockDim.x + threadIdx.x;
  if (t < n) e32[t] = (int)ei[t];
}

// s1[h] = sum_c W1[h,c]*att_src1[h,c] ; d1[h] = sum_c W1[h,c]*att_dst1[h,c]
__global__ void sd_k(const float* __restrict__ W1, const float* __restrict__ as1,
                     const float* __restrict__ ad1, float* __restrict__ sd) {
  int t = threadIdx.x;
  if (t >= 16) return;
  int h = t & 7;
  const float* att = (t < 8) ? as1 : ad1;
  float acc = 0.f;
#pragma unroll
  for (int c = 0; c < 16; ++c) acc += W1[h * 16 + c] * att[h * 16 + c];
  sd[t] = acc;
}

// ---------------- layer 1 edge passes (E real edges + N self-loop tail) ----------------

template <typename IT>
__global__ void l1_max_k(const IT* __restrict__ ei, const float* __restrict__ x,
                         const float* __restrict__ sd, unsigned* __restrict__ m1u,
                         int E, int N) {
  int t = blockIdx.x * blockDim.x + threadIdx.x;
  if (t >= E + N) return;
  int s, d;
  if (t < E) { s = (int)ei[t]; d = (int)ei[(size_t)E + t]; } else { s = d = t - E; }
  float xs = x[s], xd = x[d];
  unsigned* mp = m1u + (size_t)d * NHEAD;
#pragma unroll
  for (int h = 0; h < NHEAD; ++h) {
    float sc = lrelu(xs * sd[h] + xd * sd[NHEAD + h]);
    atomicMax(&mp[h], ordf(sc));
  }
}

template <typename IT>
__global__ void l1_sum_k(const IT* __restrict__ ei, const float* __restrict__ x,
                         const float* __restrict__ sd, const unsigned* __restrict__ m1u,
                         float* __restrict__ den1, int E, int N) {
  int t = blockIdx.x * blockDim.x + threadIdx.x;
  if (t >= E + N) return;
  int s, d;
  if (t < E) { s = (int)ei[t]; d = (int)ei[(size_t)E + t]; } else { s = d = t - E; }
  float xs = x[s], xd = x[d];
  const unsigned* mp = m1u + (size_t)d * NHEAD;
  float* dp = den1 + (size_t)d * NHEAD;
#pragma unroll
  for (int h = 0; h < NHEAD; ++h) {
    float sc = lrelu(xs * sd[h] + xd * sd[NHEAD + h]);
    atomicAdd(&dp[h], expf(sc - unordf(mp[h])));
  }
}

template <typename IT>
__global__ void l1_agg_k(const IT* __restrict__ ei, const float* __restrict__ x,
                         const float* __restrict__ sd, const unsigned* __restrict__ m1u,
                         const float* __restrict__ den1, float* __restrict__ t1,
                         int E, int N) {
  int t = blockIdx.x * blockDim.x + threadIdx.x;
  if (t >= E + N) return;
  int s, d;
  if (t < E) { s = (int)ei[t]; d = (int)ei[(size_t)E + t]; } else { s = d = t - E; }
  float xs = x[s], xd = x[d];
  const unsigned* mp = m1u + (size_t)d * NHEAD;
  const float* dp = den1 + (size_t)d * NHEAD;
  float* tp = t1 + (size_t)d * NHEAD;
#pragma unroll
  for (int h = 0; h < NHEAD; ++h) {
    float sc = lrelu(xs * sd[h] + xd * sd[NHEAD + h]);
    float alpha = expf(sc - unordf(mp[h])) / (dp[h] + EPS_DEN);
    atomicAdd(&tp[h], alpha * xs);
  }
}

// ---------------- fused ELU + (z1 @ W2) GEMV via V_WMMA_F32_16X16X4_F32 ----------------
// One wave32 computes 16 nodes. A (16x4 f32): lane M = lane&15, VGPR0 K = khalf,
// VGPR1 K = khalf+1, khalf = (lane>>4)*2 (ISA §7.12.2 layout). B (4x16) is the W2
// chunk broadcast over all 16 N-columns (robust to the N->lane map); K->lane map
// mirrors A. C/D: VGPR r holds rows M=r (lanes 0-15) and M=r+8 (lanes 16-31).
// W1/b1/W2 are staged in LDS (broadcast reads, no bank conflicts).
__global__ void h2_k(const float* __restrict__ t1, const float* __restrict__ W1,
                     const float* __restrict__ b1, const float* __restrict__ W2,
                     float* __restrict__ h2, int N) {
  __shared__ float sW1[128], sb1[128], sW2[128];
  if (threadIdx.x < 128) {
    sW1[threadIdx.x] = W1[threadIdx.x];
    sb1[threadIdx.x] = b1[threadIdx.x];
    sW2[threadIdx.x] = W2[threadIdx.x];
  }
  __syncthreads();

  int wave = blockIdx.x * (blockDim.x >> 5) + (threadIdx.x >> 5);
  int lane = threadIdx.x & 31;
  int base = wave * 16;
  if (base >= N) return;                 // wave-uniform: EXEC all-ones for WMMA
  int m = lane & 15;
  int node = base + m;
  if (node >= N) node = N - 1;           // clamp loads (N%16==0 in practice)
  int khalf = (lane >> 4) << 1;          // 0 or 2

  v8f c = {};                            // f32 16x16 accumulator
  for (int h = 0; h < NHEAD; ++h) {
    float th = t1[(size_t)node * NHEAD + h];
#pragma unroll
    for (int kk = 0; kk < 16; kk += 4) {
      int K = h * 16 + kk + khalf;       // lane's K for VGPR0 (K+1 for VGPR1)
      float a0 = eluf(th * sW1[K] + sb1[K]);
      float a1 = eluf(th * sW1[K + 1] + sb1[K + 1]);
      v2f A; A[0] = a0; A[1] = a1;
      v2f B; B[0] = sW2[K]; B[1] = sW2[K + 1];   // broadcast over N-columns
      c = __builtin_amdgcn_wmma_f32_16x16x4_f32(
          /*neg_a=*/false, A, /*neg_b=*/false, B,
          /*c_mod=*/(short)0, c, /*reuse_a=*/false, /*reuse_b=*/false);
    }
  }

  // lanes 0-7 write nodes base+0..7 from c[lane]; lanes 16-23 write nodes
  // base+8..15 from c[lane-16]; all N-columns identical.
  int r = lane & 7;
  float val;
  switch (r) {
    case 0: val = c[0]; break; case 1: val = c[1]; break;
    case 2: val = c[2]; break; case 3: val = c[3]; break;
    case 4: val = c[4]; break; case 5: val = c[5]; break;
    case 6: val = c[6]; break; default: val = c[7]; break;
  }
  if ((lane & 15) < 8) {
    int outn = base + r + ((lane >> 4) << 3);
    if (outn < N) h2[outn] = val;
  }
}

// ---------------- layer 2 edge passes (scalar feature) ----------------

template <typename IT>
__global__ void l2_max_k(const IT* __restrict__ ei, const float* __restrict__ h2,
                         const float* __restrict__ as2, const float* __restrict__ ad2,
                         unsigned* __restrict__ m2u, int E, int N) {
  int t = blockIdx.x * blockDim.x + threadIdx.x;
  if (t >= E + N) return;
  int s, d;
  if (t < E) { s = (int)ei[t]; d = (int)ei[(size_t)E + t]; } else { s = d = t - E; }
  float sc = lrelu(h2[s] * as2[0] + h2[d] * ad2[0]);
  atomicMax(&m2u[d], ordf(sc));
}

template <typename IT>
__global__ void l2_sum_k(const IT* __restrict__ ei, const float* __restrict__ h2,
                         const float* __restrict__ as2, const float* __restrict__ ad2,
                         const unsigned* __restrict__ m2u, float* __restrict__ den2,
                         int E, int N) {
  int t = blockIdx.x * blockDim.x + threadIdx.x;
  if (t >= E + N) return;
  int s, d;
  if (t < E) { s = (int)ei[t]; d = (int)ei[(size_t)E + t]; } else { s = d = t - E; }
  float sc = lrelu(h2[s] * as2[0] + h2[d] * ad2[0]);
  atomicAdd(&den2[d], expf(sc - unordf(m2u[d])));
}

template <typename IT>
__global__ void l2_agg_k(const IT* __restrict__ ei, const float* __restrict__ h2,
                         const float* __restrict__ as2, const float* __restrict__ ad2,
                         const unsigned* __restrict__ m2u, const float* __restrict__ den2,
                         float* __restrict__ acc, int E, int N) {
  int t = blockIdx.x * blockDim.x + threadIdx.x;
  if (t >= E + N) return;
  int s, d;
  if (t < E) { s = (int)ei[t]; d = (int)ei[(size_t)E + t]; } else { s = d = t - E; }
  float hs = h2[s];
  float sc = lrelu(hs * as2[0] + h2[d] * ad2[0]);
  float alpha = expf(sc - unordf(m2u[d])) / (den2[d] + EPS_DEN);
  atomicAdd(&acc[d], alpha * hs);
}

__global__ void fin_k(const float* __restrict__ acc, const float* __restrict__ b2,
                      float* __restrict__ out, int N) {
  int t = blockIdx.x * blockDim.x + threadIdx.x;
  if (t < N) out[t] = acc[t] + b2[0];
}

// ---------------- launch ----------------

extern "C" void kernel_launch(void* const* d_in, const int* in_sizes, int n_in,
                              void* d_out, int out_size, void* d_ws, size_t ws_size,
                              hipStream_t stream) {
  const float*     x   = (const float*)d_in[0];
  const long long* ei  = (const long long*)d_in[1];   // int64 [2,E]
  const float*     W1  = (const float*)d_in[2];
  const float*     as1 = (const float*)d_in[3];
  const float*     ad1 = (const float*)d_in[4];
  const float*     b1  = (const float*)d_in[5];
  const float*     W2  = (const float*)d_in[6];
  const float*     as2 = (const float*)d_in[7];
  const float*     ad2 = (const float*)d_in[8];
  const float*     b2  = (const float*)d_in[9];

  const int N = in_sizes[0];          // x is [N,1]
  const int E = in_sizes[1] / 2;      // edge_index is [2,E]

  // workspace (floats): t1[8N] den1[8N] m1u[8N] h2[N] den2[N] m2u[N] acc[N] sd[16] | e32[2E]
  float*    ws   = (float*)d_ws;
  size_t    n8   = (size_t)N * NHEAD;
  float*    t1   = ws;
  float*    den1 = ws + n8;
  unsigned* m1u  = (unsigned*)(ws + 2 * n8);
  float*    h2   = ws + 3 * n8;
  float*    den2 = h2 + N;
  unsigned* m2u  = (unsigned*)(den2 + N);
  float*    acc  = (float*)(m2u + N);
  float*    sd   = acc + N;
  size_t    ztot = 28 * (size_t)N + 16;
  int*      e32  = (int*)(ws + ztot);
  bool use32 = ws_size >= ztot * 4 + (size_t)2 * E * 4;

  const int B  = 256;
  const int EG = (E + N + B - 1) / B;

  zero_k<<<(int)((ztot + B - 1) / B), B, 0, stream>>>((unsigned*)d_ws, ztot);
  sd_k<<<1, 32, 0, stream>>>(W1, as1, ad1, sd);
  if (use32)
    cvt_k<<<(int)(((size_t)2 * E + B - 1) / B), B, 0, stream>>>(ei, e32, (size_t)2 * E);

  if (use32) {
    l1_max_k<<<EG, B, 0, stream>>>(e32, x, sd, m1u, E, N);
    l1_sum_k<<<EG, B, 0, stream>>>(e32, x, sd, m1u, den1, E, N);
    l1_agg_k<<<EG, B, 0, stream>>>(e32, x, sd, m1u, den1, t1, E, N);
  } else {
    l1_max_k<<<EG, B, 0, stream>>>(ei, x, sd, m1u, E, N);
    l1_sum_k<<<EG, B, 0, stream>>>(ei, x, sd, m1u, den1, E, N);
    l1_agg_k<<<EG, B, 0, stream>>>(ei, x, sd, m1u, den1, t1, E, N);
  }

  int tiles = (N + 15) / 16;
  int wpb   = B / 32;  // 8 waves/block, one 16-node tile per wave
  h2_k<<<(tiles + wpb - 1) / wpb, B, 0, stream>>>(t1, W1, b1, W2, h2, N);

  if (use32) {
    l2_max_k<<<EG, B, 0, stream>>>(e32, h2, as2, ad2, m2u, E, N);
    l2_sum_k<<<EG, B, 0, stream>>>(e32, h2, as2, ad2, m2u, den2, E, N);
    l2_agg_k<<<EG, B, 0, stream>>>(e32, h2, as2, ad2, m2u, den2, acc, E, N);
  } else {
    l2_max_k<<<EG, B, 0, stream>>>(ei, h2, as2, ad2, m2u, E, N);
    l2_sum_k<<<EG, B, 0, stream>>>(ei, h2, as2, ad2, m2u, den2, E, N);
    l2_agg_k<<<EG, B, 0, stream>>>(ei, h2, as2, ad2, m2u, den2, acc, E, N);
  }

  fin_k<<<(N + B - 1) / B, B, 0, stream>>>(acc, b2, (float*)d_out, N);
}